// VINDyLayer_14534169329675
// MI455X (gfx1250) — compile-verified
//
#include <hip/hip_runtime.h>

// VINDy layer: out[B,18] = Theta(x)[B,1330] @ (Xi*mask)[1330,18], x=[z|betas], B=131072.
// Compute-bound fp32 (AI ~350 FLOP/B) -> V_WMMA_F32_16X16X4_F32.
// A-fragments (Theta) are synthesized in straight-line VALU from a constexpr monomial
// table (fully unrolled 333 K-groups); B-fragments are immediate-offset ds_load_b64
// from a transposed, zero-padded Xi*mask image in LDS.

#define NV 18
#define NP 1330
#define KP 1332              // NP padded to a multiple of 4 (pad rows of Xi are zero)
#define NCOLS 20             // 18 real columns + 2 zero columns for the N=16..31 tail
#define NGROUPS (KP / 4)     // 333
#define WAVES 8
#define ROWS_PER_WAVE 16
#define ROWS_PER_BLOCK (WAVES * ROWS_PER_WAVE)   // 128

typedef __attribute__((ext_vector_type(2))) float v2f;
typedef __attribute__((ext_vector_type(4))) float v4f;
typedef __attribute__((ext_vector_type(8))) float v8f;

// ---- compile-time monomial table (reference feature order, homogenized) ----
struct MonTab {
    unsigned char a[KP];
    unsigned char b[KP];
    unsigned char c[KP];
};
constexpr MonTab make_tab() {
    MonTab t{};
    int f = 0;
    t.a[f] = 18; t.b[f] = 18; t.c[f] = 18; ++f;                       // constant term
    for (int i = 0; i < NV; ++i) { t.a[f] = i; t.b[f] = 18; t.c[f] = 18; ++f; }
    for (int i = 0; i < NV; ++i)
        for (int j = i; j < NV; ++j) { t.a[f] = i; t.b[f] = j; t.c[f] = 18; ++f; }
    for (int i = 0; i < NV; ++i)
        for (int j = i; j < NV; ++j)
            for (int k = j; k < NV; ++k) { t.a[f] = i; t.b[f] = j; t.c[f] = k; ++f; }
    for (; f < KP; ++f) { t.a[f] = 18; t.b[f] = 18; t.c[f] = 18; }    // pads (B rows are 0)
    return t;
}
constexpr MonTab TAB = make_tab();

// monomial F for one row held in registers; x[18] == 1.0 slots fold away
template <int F>
static __device__ __forceinline__ float mono(const float* x) {
    constexpr int ia = TAB.a[F];
    constexpr int ib = TAB.b[F];
    constexpr int ic = TAB.c[F];
    float p = x[ia];
    if constexpr (ib != 18) p *= x[ib];
    if constexpr (ic != 18) p *= x[ic];
    return p;
}

template <int G, int GEnd>
struct Steps {
    static __device__ __forceinline__ void run(const float* x, bool hi,
                                               const float* b0p, const float* b1p,
                                               v8f& acc0, v8f& acc1) {
        constexpr int f = 4 * G;
        // 4 features of this group; lane keeps its half's pair
        float p0 = mono<f + 0>(x);
        float p1 = mono<f + 1>(x);
        float p2 = mono<f + 2>(x);
        float p3 = mono<f + 3>(x);
        v2f A;
        A.x = hi ? p2 : p0;
        A.y = hi ? p3 : p1;
        // B fragments: one 8-byte LDS load each, immediate offset (16*G bytes)
        v2f B0 = *(const v2f*)(b0p + f);
        v2f B1 = *(const v2f*)(b1p + f);
        acc0 = __builtin_amdgcn_wmma_f32_16x16x4_f32(
                   false, A, false, B0, (short)0, acc0, false, false);
        acc1 = __builtin_amdgcn_wmma_f32_16x16x4_f32(
                   false, A, false, B1, (short)0, acc1, false, false);
        Steps<G + 1, GEnd>::run(x, hi, b0p, b1p, acc0, acc1);
    }
};
template <int GEnd>
struct Steps<GEnd, GEnd> {
    static __device__ __forceinline__ void run(const float*, bool, const float*,
                                               const float*, v8f&, v8f&) {}
};

__global__ __launch_bounds__(256)
void vindy_wmma_kernel(const float* __restrict__ z,
                       const float* __restrict__ betas,
                       const float* __restrict__ big_xi,
                       const float* __restrict__ mask,
                       float* __restrict__ out,
                       int batch)
{
    // Xi*mask, transposed (column-major over K), K padded to KP, cols 18/19 zero.
    __shared__ float sXiT[NCOLS][KP];   // 20*1332*4 = 106,560 B

    const int tid  = threadIdx.x;
    const int lane = tid & 31;
    const int w    = tid >> 5;
    const int half = lane >> 4;        // 0: lanes 0-15, 1: lanes 16-31
    const int lm   = lane & 15;
    const int rowBase = blockIdx.x * ROWS_PER_BLOCK + w * ROWS_PER_WAVE;

    // --- stage Xi*mask into LDS (one-time, L2-resident source) ---
    for (int col = 0; col < NCOLS; ++col) {
        for (int k = tid; k < KP; k += 256) {
            float v = 0.0f;
            if (col < NV && k < NP) v = big_xi[k * NV + col] * mask[k * NV + col];
            sXiT[col][k] = v;
        }
    }

    // --- this lane's row of x = [z(16) | betas(2) | 1.0], register resident ---
    int gr = rowBase + lm;
    if (gr > batch - 1) gr = batch - 1;
    float x[19];
    {
        const v4f* zr = (const v4f*)(z + (size_t)gr * 16);
        v4f q0 = zr[0], q1 = zr[1], q2 = zr[2], q3 = zr[3];
        x[0]  = q0.x; x[1]  = q0.y; x[2]  = q0.z; x[3]  = q0.w;
        x[4]  = q1.x; x[5]  = q1.y; x[6]  = q1.z; x[7]  = q1.w;
        x[8]  = q2.x; x[9]  = q2.y; x[10] = q2.z; x[11] = q2.w;
        x[12] = q3.x; x[13] = q3.y; x[14] = q3.z; x[15] = q3.w;
        const v2f* br = (const v2f*)(betas + (size_t)gr * 2);
        v2f bb = br[0];
        x[16] = bb.x; x[17] = bb.y; x[18] = 1.0f;
    }

    __syncthreads();

    v8f acc0 = {0.f, 0.f, 0.f, 0.f, 0.f, 0.f, 0.f, 0.f};  // out cols 0..15
    v8f acc1 = {0.f, 0.f, 0.f, 0.f, 0.f, 0.f, 0.f, 0.f};  // out cols 16..17

    const bool hi = (half != 0);
    const float* b0p = &sXiT[lm][2 * half];                         // tile0 column
    const int col1   = (lm < 2) ? (16 + lm) : 18;                   // tile1: zeros if col>=18
    const float* b1p = &sXiT[col1][2 * half];

    Steps<0, NGROUPS>::run(x, hi, b0p, b1p, acc0, acc1);

    // --- D layout: lane holds out[row v+8*half][col lm] in acc[v] ---
    const int orow = rowBase + 8 * half;
    if (rowBase + ROWS_PER_WAVE <= batch) {                         // uniform fast path
#pragma unroll
        for (int v = 0; v < 8; ++v)
            out[(size_t)(orow + v) * NV + lm] = acc0[v];
        if (lm < 2) {
#pragma unroll
            for (int v = 0; v < 8; ++v)
                out[(size_t)(orow + v) * NV + 16 + lm] = acc1[v];
        }
    } else {
#pragma unroll
        for (int v = 0; v < 8; ++v) {
            int r = orow + v;
            if (r < batch) {
                out[(size_t)r * NV + lm] = acc0[v];
                if (lm < 2) out[(size_t)r * NV + 16 + lm] = acc1[v];
            }
        }
    }
}

extern "C" void kernel_launch(void* const* d_in, const int* in_sizes, int n_in,
                              void* d_out, int out_size, void* d_ws, size_t ws_size,
                              hipStream_t stream) {
    const float* z      = (const float*)d_in[0];   // [B,16]
    const float* betas  = (const float*)d_in[1];   // [B,2]
    const float* big_xi = (const float*)d_in[2];   // [1330,18]
    const float* mask   = (const float*)d_in[3];   // [1330,18]
    float* out          = (float*)d_out;           // [B,18]

    const int batch = in_sizes[0] / 16;            // 131072
    const int grid  = (batch + ROWS_PER_BLOCK - 1) / ROWS_PER_BLOCK;

    vindy_wmma_kernel<<<grid, 256, 0, stream>>>(z, betas, big_xi, mask, out, batch);
}